// Synthesis3dResBlock_35665408426091
// MI455X (gfx1250) — compile-verified
//
#include <hip/hip_runtime.h>
#include <math.h>

typedef __attribute__((ext_vector_type(2))) float v2f;
typedef __attribute__((ext_vector_type(8))) float v8f;

#define N_  4
#define CI_ 128
#define CO_ 128
#define Z_  512
#define T0_ 8
#define H0_ 16
#define W0_ 16
#define T1_ 16
#define H1_ 32
#define W1_ 32

// ---------------- separable 2x linear upsample: y[2i]=.25x[i-1]+.75x[i], y[2i+1]=.75x[i]+.25x[i+1]
__global__ void upsample1d_kernel(const float* __restrict__ src, float* __restrict__ dst,
                                  int outer, int L, int inner) {
  int total = outer * 2 * L * inner;
  int id = blockIdx.x * blockDim.x + threadIdx.x;
  if (id >= total) return;
  int in_i = id % inner;
  int j = (id / inner) % (2 * L);
  int o = id / (inner * 2 * L);
  int i = j >> 1;
  const float* s = src + (size_t)o * L * inner + in_i;
  float v;
  if ((j & 1) == 0) {
    float a = (i - 1 >= 0) ? s[(size_t)(i - 1) * inner] : 0.f;
    float b = s[(size_t)i * inner];
    v = 0.25f * a + 0.75f * b;
  } else {
    float a = s[(size_t)i * inner];
    float b = (i + 1 < L) ? s[(size_t)(i + 1) * inner] : 0.f;
    v = 0.75f * a + 0.25f * b;
  }
  dst[id] = v;
}

// ---------------- style = (1/sqrt(Z)) * latent^T W + b     (layout: style[(n*CI+c)*Tn+t])
__global__ void affine_kernel(const float* __restrict__ lat, const float* __restrict__ Wm,
                              const float* __restrict__ bias, float* __restrict__ style,
                              int Tn, float gain) {
  int total = N_ * CI_ * Tn;
  int id = blockIdx.x * blockDim.x + threadIdx.x;
  if (id >= total) return;
  int t = id % Tn;
  int c = (id / Tn) % CI_;
  int n = id / (Tn * CI_);
  const float* lp = lat + (size_t)n * Z_ * Tn + t;
  const float* wp = Wm + (size_t)c * Z_;
  float s = 0.f;
  for (int z = 0; z < Z_; ++z) s += lp[(size_t)z * Tn] * wp[z];
  style[id] = s * gain + bias[c];
}

// ---------------- per-batch normalize style by max|style|
__global__ void stylenorm_kernel(const float* __restrict__ style, float* __restrict__ stylen, int CT) {
  int n = blockIdx.x;
  int tid = threadIdx.x;
  __shared__ float red[256];
  const float* s = style + (size_t)n * CT;
  float m = 0.f;
  for (int i = tid; i < CT; i += 256) m = fmaxf(m, fabsf(s[i]));
  red[tid] = m;
  __syncthreads();
  for (int st = 128; st > 0; st >>= 1) {
    if (tid < st) red[tid] = fmaxf(red[tid], red[tid + st]);
    __syncthreads();
  }
  float inv = 1.f / red[0];
  for (int i = tid; i < CT; i += 256) stylen[(size_t)n * CT + i] = s[i] * inv;
}

// ---------------- wn = w / (max|w|_o * sqrt(Ci*27));  wsq[o,i] = sum_taps wn^2
__global__ void prepw_kernel(const float* __restrict__ w, float* __restrict__ wn,
                             float* __restrict__ wsq) {
  const int K = CI_ * 27;
  int o = blockIdx.x;
  int tid = threadIdx.x; // 128 threads
  __shared__ float red[128];
  const float* wo = w + (size_t)o * K;
  float m = 0.f;
  for (int i = tid; i < K; i += 128) m = fmaxf(m, fabsf(wo[i]));
  red[tid] = m;
  __syncthreads();
  for (int st = 64; st > 0; st >>= 1) {
    if (tid < st) red[tid] = fmaxf(red[tid], red[tid + st]);
    __syncthreads();
  }
  float scale = 1.f / (red[0] * sqrtf((float)K));
  for (int i = tid; i < K; i += 128) wn[(size_t)o * K + i] = wo[i] * scale;
  const float* wi = wo + (size_t)tid * 27;
  float s = 0.f;
  for (int k = 0; k < 27; ++k) { float v = wi[k] * scale; s += v * v; }
  wsq[o * CI_ + tid] = s;
}

// ---------------- demod[n,o,t] = rsqrt( sum_i wsq[o,i]*stylen[n,i,t]^2 + 1e-8 )
__global__ void demod_kernel(const float* __restrict__ wsq, const float* __restrict__ stylen,
                             float* __restrict__ demod, int Tn) {
  int total = N_ * CO_ * Tn;
  int id = blockIdx.x * blockDim.x + threadIdx.x;
  if (id >= total) return;
  int t = id % Tn;
  int o = (id / Tn) % CO_;
  int n = id / (Tn * CO_);
  const float* sp = stylen + (size_t)n * CI_ * Tn + t;
  const float* wq = wsq + (size_t)o * CI_;
  float s = 0.f;
  for (int i = 0; i < CI_; ++i) { float sv = sp[(size_t)i * Tn]; s += wq[i] * sv * sv; }
  demod[id] = rsqrtf(s + 1e-8f);
}

// ---------------- xm = x * stylen[n,c,t]
__global__ void modulate_kernel(const float* __restrict__ x, const float* __restrict__ stylen,
                                float* __restrict__ xm, int Tn, int HWn) {
  int total = N_ * CI_ * Tn * HWn;
  int id = blockIdx.x * blockDim.x + threadIdx.x;
  if (id >= total) return;
  int t = (id / HWn) % Tn;
  int c = (id / (HWn * Tn)) % CI_;
  int n = id / (HWn * Tn * CI_);
  xm[id] = x[id] * stylen[(n * CI_ + c) * Tn + t];
}

// ---------------- skip: out[n,co,t,h,w] = sum_ci (wsk[co,ci]/sqrt(Ci)) * x[n,ci,t,h,w]  (WMMA f32)
__global__ __launch_bounds__(256) void skipgemm_kernel(const float* __restrict__ x,
                                                       const float* __restrict__ wsk,
                                                       float* __restrict__ out) {
  int n = blockIdx.z / T0_, t = blockIdx.z % T0_;
  int h = blockIdx.y;
  int w0 = blockIdx.x * 16;
  int tid = threadIdx.x, lane = tid & 31, wave = tid >> 5;
  __shared__ float bl[CI_ * 16];
  for (int e = tid; e < CI_ * 16; e += 256) {
    int ci = e >> 4, ww = e & 15;
    bl[e] = x[(((size_t)(n * CI_ + ci) * T0_ + t) * H0_ + h) * W0_ + w0 + ww];
  }
  __syncthreads();
  const float sc = 0.08838834764831845f; // 1/sqrt(128)
  int m = lane & 15, kk = (lane >> 4) << 1;
  int cob = wave * 16;
  v8f acc = {0.f, 0.f, 0.f, 0.f, 0.f, 0.f, 0.f, 0.f};
  for (int c4 = 0; c4 < CI_; c4 += 4) {
    int ci = c4 + kk;
    v2f a, b;
    a.x = wsk[(size_t)(cob + m) * CI_ + ci] * sc;
    a.y = wsk[(size_t)(cob + m) * CI_ + ci + 1] * sc;
    b.x = bl[ci * 16 + m];
    b.y = bl[(ci + 1) * 16 + m];
    acc = __builtin_amdgcn_wmma_f32_16x16x4_f32(false, a, false, b, (short)0, acc, false, false);
  }
  for (int r = 0; r < 8; ++r) {
    int co = cob + r + ((lane >> 4) << 3);
    int ww = w0 + (lane & 15);
    out[(((size_t)(n * CO_ + co) * T0_ + t) * H0_ + h) * W0_ + ww] = acc[r];
  }
}

// ---------------- 3x3x3 modulated conv as implicit GEMM on WMMA f32 16x16x4.
// One block = one (n,t,h, 16-wide w tile), 8 waves = 8 co-tiles of 16.
// LDS stages 16 ci x 3t x 3h x 18w modulated-input patch per K chunk.
template <int FINAL>
__global__ __launch_bounds__(256) void conv3d_wmma_kernel(
    const float* __restrict__ xm, const float* __restrict__ wn,
    const float* __restrict__ demod, const float* __restrict__ bias,
    const float* __restrict__ skip, float* __restrict__ out,
    int Tn, int Hn, int Wn) {
  int n = blockIdx.z / Tn, t = blockIdx.z % Tn;
  int h = blockIdx.y;
  int w0 = blockIdx.x * 16;
  int tid = threadIdx.x, lane = tid & 31, wave = tid >> 5;
  int cob = wave * 16;
  const int HW = Hn * Wn;
  __shared__ float lds[16 * 3 * 3 * 18]; // [ci_l][dt][dh][wi]: stride 162 per ci_l
  v8f acc = {0.f, 0.f, 0.f, 0.f, 0.f, 0.f, 0.f, 0.f};
  int m = lane & 15, kk = (lane >> 4) << 1;
  for (int ci0 = 0; ci0 < CI_; ci0 += 16) {
    __syncthreads();
    for (int e = tid; e < 16 * 162; e += 256) {
      int ci_l = e / 162, r = e % 162;
      int dt = r / 54, r2 = r % 54, dh = r2 / 18, wi = r2 % 18;
      int tt = t + dt - 1, hh = h + dh - 1, ww = w0 + wi - 1;
      float v = 0.f;
      if (tt >= 0 && tt < Tn && hh >= 0 && hh < Hn && ww >= 0 && ww < Wn)
        v = xm[((size_t)(n * CI_ + ci0 + ci_l) * Tn + tt) * HW + hh * Wn + ww];
      lds[e] = v;
    }
    if (ci0 + 16 < CI_) {
      const float* pf = xm + ((size_t)(n * CI_ + ci0 + 16 + (tid & 15)) * Tn + t) * HW +
                        (size_t)h * Wn + w0;
      __builtin_prefetch(pf, 0, 1);
    }
    __syncthreads();
    for (int tap = 0; tap < 27; ++tap) {
      int dt = tap / 9, dh = (tap % 9) / 3, dw = tap % 3;
      int bbase = dt * 54 + dh * 18 + m + dw;
      for (int j = 0; j < 4; ++j) {
        int ci = ci0 + j * 4 + kk;
        const float* wp = wn + ((size_t)(cob + m) * CI_ + ci) * 27 + tap;
        v2f a, b;
        a.x = wp[0];
        a.y = wp[27];
        int cl = (j * 4 + kk) * 162 + bbase;
        b.x = lds[cl];
        b.y = lds[cl + 162];
        acc = __builtin_amdgcn_wmma_f32_16x16x4_f32(false, a, false, b, (short)0, acc, false, false);
      }
    }
  }
  const float g = 1.41421356237309515f;
  for (int r = 0; r < 8; ++r) {
    int co = cob + r + ((lane >> 4) << 3);
    int ww = w0 + (lane & 15);
    float v = acc[r] * demod[(n * CO_ + co) * Tn + t] + bias[co];
    v = (v >= 0.f ? v : 0.2f * v) * g;
    v = fminf(fmaxf(v, -256.f), 256.f);
    size_t oi = ((size_t)(n * CO_ + co) * Tn + t) * HW + (size_t)h * Wn + ww;
    if (FINAL) v = (v + skip[oi]) * 0.70710678118654746f;
    out[oi] = v;
  }
}

extern "C" void kernel_launch(void* const* d_in, const int* in_sizes, int n_in,
                              void* d_out, int out_size, void* d_ws, size_t ws_size,
                              hipStream_t stream) {
  (void)in_sizes; (void)n_in; (void)out_size; (void)ws_size;
  const float* input  = (const float*)d_in[0];
  const float* latent = (const float*)d_in[1];
  const float* aff0_w = (const float*)d_in[2];
  const float* aff0_b = (const float*)d_in[3];
  const float* aff1_w = (const float*)d_in[4];
  const float* aff1_b = (const float*)d_in[5];
  const float* w0     = (const float*)d_in[6];
  const float* w1     = (const float*)d_in[7];
  const float* w_skip = (const float*)d_in[8];
  const float* b0     = (const float*)d_in[9];
  const float* b1     = (const float*)d_in[10];
  float* out = (float*)d_out;
  float* ws  = (float*)d_ws;

  // small-region offsets (floats)
  float* style0  = ws + 0;        // 4096
  float* stylen0 = ws + 4096;     // 4096
  float* demod0  = ws + 8192;     // 4096
  float* latup   = ws + 12288;    // 32768
  float* style1  = ws + 45056;    // 8192
  float* stylen1 = ws + 53248;    // 8192
  float* demod1  = ws + 61440;    // 8192
  float* w0n     = ws + 69632;    // 442368
  float* wsq0    = ws + 512000;   // 16384
  float* w1n     = ws + 528384;   // 442368
  float* wsq1    = ws + 970752;   // 16384
  float* P0 = ws + (1u << 20);    // 1M : xmod0, later skip0
  float* P1 = P0 + (1u << 20);    // 1M : x0act
  float* P2 = P1 + (1u << 20);    // 2M : T-upsampled
  float* P3 = P2 + (2u << 20);    // 4M : TH-upsampled
  float* P4 = P3 + (4u << 20);    // 8M : full-res x, later full-res skip
  float* P5 = P4 + (8u << 20);    // 8M : xmod1

  const float gainA = 0.04419417382415922f; // 1/sqrt(512)

  // latent temporal upsample (N,Z,8)->(N,Z,16)
  { int tot = N_ * Z_ * T1_;
    upsample1d_kernel<<<(tot + 255) / 256, 256, 0, stream>>>(latent, latup, N_ * Z_, T0_, 1); }
  // affines
  { int tot = N_ * CI_ * T0_;
    affine_kernel<<<(tot + 255) / 256, 256, 0, stream>>>(latent, aff0_w, aff0_b, style0, T0_, gainA); }
  { int tot = N_ * CI_ * T1_;
    affine_kernel<<<(tot + 255) / 256, 256, 0, stream>>>(latup, aff1_w, aff1_b, style1, T1_, gainA); }
  // style normalization
  stylenorm_kernel<<<N_, 256, 0, stream>>>(style0, stylen0, CI_ * T0_);
  stylenorm_kernel<<<N_, 256, 0, stream>>>(style1, stylen1, CI_ * T1_);
  // weight prep
  prepw_kernel<<<CO_, 128, 0, stream>>>(w0, w0n, wsq0);
  prepw_kernel<<<CO_, 128, 0, stream>>>(w1, w1n, wsq1);
  // demodulation coefficients
  { int tot = N_ * CO_ * T0_;
    demod_kernel<<<(tot + 255) / 256, 256, 0, stream>>>(wsq0, stylen0, demod0, T0_); }
  { int tot = N_ * CO_ * T1_;
    demod_kernel<<<(tot + 255) / 256, 256, 0, stream>>>(wsq1, stylen1, demod1, T1_); }
  // xmod0 = input * stylen0
  { int tot = N_ * CI_ * T0_ * H0_ * W0_;
    modulate_kernel<<<(tot + 255) / 256, 256, 0, stream>>>(input, stylen0, P0, T0_, H0_ * W0_); }
  // conv0 (+demod+bias+lrelu+clamp) -> P1
  { dim3 g(W0_ / 16, H0_, N_ * T0_);
    conv3d_wmma_kernel<0><<<g, 256, 0, stream>>>(P0, w0n, demod0, b0, nullptr, P1, T0_, H0_, W0_); }
  // skip 1x1x1 conv -> P0 (xmod0 dead now)
  { dim3 g(W0_ / 16, H0_, N_ * T0_);
    skipgemm_kernel<<<g, 256, 0, stream>>>(input, w_skip, P0); }
  // x upsample chain: P1 -T-> P2 -H-> P3 -W-> P4
  { int tot = N_ * CI_ * T1_ * H0_ * W0_;
    upsample1d_kernel<<<(tot + 255) / 256, 256, 0, stream>>>(P1, P2, N_ * CI_, T0_, H0_ * W0_); }
  { int tot = N_ * CI_ * T1_ * H1_ * W0_;
    upsample1d_kernel<<<(tot + 255) / 256, 256, 0, stream>>>(P2, P3, N_ * CI_ * T1_, H0_, W0_); }
  { int tot = N_ * CI_ * T1_ * H1_ * W1_;
    upsample1d_kernel<<<(tot + 255) / 256, 256, 0, stream>>>(P3, P4, N_ * CI_ * T1_ * H1_, W0_, 1); }
  // xmod1 = xfull * stylen1 -> P5  (P4 free afterward)
  { int tot = N_ * CI_ * T1_ * H1_ * W1_;
    modulate_kernel<<<(tot + 255) / 256, 256, 0, stream>>>(P4, stylen1, P5, T1_, H1_ * W1_); }
  // skip upsample chain: P0 -T-> P2 -H-> P3 -W-> P4
  { int tot = N_ * CO_ * T1_ * H0_ * W0_;
    upsample1d_kernel<<<(tot + 255) / 256, 256, 0, stream>>>(P0, P2, N_ * CO_, T0_, H0_ * W0_); }
  { int tot = N_ * CO_ * T1_ * H1_ * W0_;
    upsample1d_kernel<<<(tot + 255) / 256, 256, 0, stream>>>(P2, P3, N_ * CO_ * T1_, H0_, W0_); }
  { int tot = N_ * CO_ * T1_ * H1_ * W1_;
    upsample1d_kernel<<<(tot + 255) / 256, 256, 0, stream>>>(P3, P4, N_ * CO_ * T1_ * H1_, W0_, 1); }
  // conv1 (+demod+bias+lrelu+clamp) fused with (x+skip)*sqrt(0.5) -> out
  { dim3 g(W1_ / 16, H1_, N_ * T1_);
    conv3d_wmma_kernel<1><<<g, 256, 0, stream>>>(P5, w1n, demod1, b1, P4, out, T1_, H1_, W1_); }
}